// MLLAttention_19215683682790
// MI455X (gfx1250) — compile-verified
//
#include <hip/hip_runtime.h>
#include <math.h>

typedef __attribute__((ext_vector_type(2))) float v2f;
typedef __attribute__((ext_vector_type(4))) float v4f;
typedef __attribute__((ext_vector_type(8))) float v8f;
typedef int vi4 __attribute__((vector_size(16)));          // int4 vector
typedef vi4 __attribute__((address_space(1))) *gvi4_p;      // global int4*
typedef vi4 __attribute__((address_space(3))) *lvi4_p;      // LDS int4*

#define B_    8
#define C_    128
#define H_    160
#define W_    160
#define N_    (H_ * W_)
#define NH    8
#define HD    16
#define TWO_C 256
#define SEG   25    // kv reduction segments per (b,h)
#define ASTR  132   // padded LDS row stride (dwords): rows 4 banks apart

#if defined(__has_builtin)
#if __has_builtin(__builtin_amdgcn_global_load_async_to_lds_b128)
#define HAVE_ASYNC_LDS 1
#endif
#if __has_builtin(__builtin_amdgcn_s_wait_asynccnt)
#define HAVE_WAIT_ASYNC 1
#endif
#endif

__device__ __forceinline__ float elu1(float x) {
    return x > 0.f ? x + 1.f : __expf(x);
}

__device__ __forceinline__ void wait_asynccnt0() {
#if defined(HAVE_WAIT_ASYNC)
    __builtin_amdgcn_s_wait_asynccnt(0);
#else
    asm volatile("s_wait_asynccnt 0x0" ::: "memory");
#endif
}

// ---------------------------------------------------------------------------
// K1: qk = xf @ qk_w.T + qk_b ; q = elu(qk[:, :128])+1 ; k = elu(qk[:,128:])+1
// Block = 8 waves sharing one M-tile (tm). A tile (16x128, 8KB) is staged to
// LDS once per block via async global->LDS (ASYNCcnt path) and reused by all
// 8 waves; each wave owns one 16x16 output tile (distinct tn), K-stepped by 4
// through V_WMMA_F32_16X16X4_F32.
// ---------------------------------------------------------------------------
__global__ void __launch_bounds__(256) qk_gemm_kernel(
    const float* __restrict__ X, const float* __restrict__ Wqk,
    const float* __restrict__ bqk, float* __restrict__ qbuf,
    float* __restrict__ kbuf)
{
    __shared__ float Atile[16 * ASTR];

    const int lane = threadIdx.x & 31;
    const int wave = threadIdx.x >> 5;
    const int TN = TWO_C / 16;          // 16
    const int TM = N_ / 16;             // 1600
    const int wid0 = blockIdx.x * 8;    // waves in a block share tm and b
    const int tm = (wid0 / TN) % TM;
    const int b  = wid0 / (TN * TM);
    const int tn = (wid0 + wave) % TN;

    // ---- stage A tile (16 rows x 128 cols) into LDS: 8 floats per thread ----
    {
        const int r  = threadIdx.x >> 4;        // 0..15
        const int c0 = (threadIdx.x & 15) * 8;  // 0..120
        const float* gsrc = X + ((size_t)b * N_ + tm * 16 + r) * C_ + c0;
        float* ldst = &Atile[r * ASTR + c0];
#if defined(HAVE_ASYNC_LDS)
        __builtin_amdgcn_global_load_async_to_lds_b128(
            (gvi4_p)(gsrc), (lvi4_p)(ldst), 0, 0);
        __builtin_amdgcn_global_load_async_to_lds_b128(
            (gvi4_p)(gsrc + 4), (lvi4_p)(ldst + 4), 0, 0);
        wait_asynccnt0();
#else
        ((v4f*)ldst)[0] = ((const v4f*)gsrc)[0];
        ((v4f*)ldst)[1] = ((const v4f*)gsrc)[1];
#endif
    }
    __syncthreads();

    const int col   = tn * 16 + (lane & 15);
    const int khalf = (lane >> 4) * 2;  // lanes 0-15: K {0,1}; lanes 16-31: K {2,3}
    const float* Als  = &Atile[(lane & 15) * ASTR];
    const float* Brow = Wqk + (size_t)col * C_;      // B[k][j] = qk_w[j][k]

    v8f acc = {};
    for (int kk = 0; kk < C_; kk += 4) {
        v2f a = {}, bv = {};
        a.x  = Als[kk + khalf];
        a.y  = Als[kk + khalf + 1];
        bv.x = Brow[kk + khalf];
        bv.y = Brow[kk + khalf + 1];
        acc = __builtin_amdgcn_wmma_f32_16x16x4_f32(
            false, a, false, bv, (short)0, acc, false, false);
    }

    const float bias  = bqk[col];
    const int   mbase = tm * 16 + ((lane >> 4) << 3);  // C layout: lanes>=16 -> M+8
    float* dst  = (col < C_) ? qbuf : kbuf;
    const int dcol = (col < C_) ? col : (col - C_);
#pragma unroll
    for (int r = 0; r < 8; ++r) {
        float v = acc[r] + bias;
        dst[((size_t)b * N_ + mbase + r) * C_ + dcol] = elu1(v);
    }
}

// ---------------------------------------------------------------------------
// K2a: kmean[b,ch] = mean_n k[b,n,ch]
// ---------------------------------------------------------------------------
__global__ void __launch_bounds__(256) kmean_kernel(
    const float* __restrict__ kbuf, float* __restrict__ kmean)
{
    __shared__ float red[256];
    const int b  = blockIdx.x / C_;
    const int ch = blockIdx.x % C_;
    float s = 0.f;
    for (int i = threadIdx.x; i < N_; i += 256)
        s += kbuf[((size_t)b * N_ + i) * C_ + ch];
    red[threadIdx.x] = s;
    __syncthreads();
    for (int off = 128; off > 0; off >>= 1) {
        if ((int)threadIdx.x < off) red[threadIdx.x] += red[threadIdx.x + off];
        __syncthreads();
    }
    if (threadIdx.x == 0) kmean[blockIdx.x] = red[0] * (1.0f / (float)N_);
}

// ---------------------------------------------------------------------------
// K2b: kv[b,h,d,e] = sum_n (rope(k)[n,d]*s) * (v[n,e]*s), s = N^-0.5 = 1/160.
// WMMA with A = roped k^T fragment (built per lane), B = scaled v fragment.
// Partial 16x16 results atomically accumulated.
// ---------------------------------------------------------------------------
__global__ void __launch_bounds__(256) kv_kernel(
    const float* __restrict__ kbuf, const float* __restrict__ X,
    float* __restrict__ kvbuf)
{
    const int lane = threadIdx.x & 31;
    const int wave = threadIdx.x >> 5;
    int bid = blockIdx.x;
    const int seg = bid % SEG;
    const int h   = (bid / SEG) % NH;
    const int b   = bid / (SEG * NH);

    const float scale = 1.0f / 160.0f;

    const int dloc = lane & 15;          // A row (d) == B col (e) for this lane
    const int nsub = (lane >> 4) * 2;    // lane half selects K(=n) pair
    const int ch   = h * HD + dloc;
    const int ch0  = ch & ~1;
    const int pair = ch >> 1;
    const float theta = __powf(10000.0f, -(float)pair * (1.0f / 64.0f));
    const bool  odd   = (ch & 1) != 0;

    const int chunk = N_ / SEG / 8;      // 128 rows per wave
    const int n0 = seg * (N_ / SEG) + wave * chunk;
    const size_t base = (size_t)b * N_ * C_;

    v8f acc = {};
    for (int nn = 0; nn < chunk; nn += 4) {
        v2f a = {}, bv = {};
#pragma unroll
        for (int t = 0; t < 2; ++t) {
            const int n = n0 + nn + nsub + t;
            const float k0 = kbuf[base + (size_t)n * C_ + ch0];
            const float k1 = kbuf[base + (size_t)n * C_ + ch0 + 1];
            float s, c;
            __sincosf((float)(n % W_) * theta, &s, &c);
            const float kr = odd ? (k0 * s + k1 * c) : (k0 * c - k1 * s);
            const float vv = X[base + (size_t)n * C_ + h * HD + dloc] * scale;
            if (t == 0) { a.x = kr * scale; bv.x = vv; }
            else        { a.y = kr * scale; bv.y = vv; }
        }
        acc = __builtin_amdgcn_wmma_f32_16x16x4_f32(
            false, a, false, bv, (short)0, acc, false, false);
    }

    const int dbase = (lane >> 4) << 3;
#pragma unroll
    for (int r = 0; r < 8; ++r) {
        const int d = dbase + r;
        atomicAdd(&kvbuf[(((size_t)b * NH + h) * HD + d) * HD + (lane & 15)],
                  acc[r]);
    }
}

// ---------------------------------------------------------------------------
// K3: out[b,ch,n] = z * (q_rope[n,h,:] . kv[b,h,:,e]) + LePE(v)[b,ch,n]
// One block per (n, b); 128 threads = one per channel.
// ---------------------------------------------------------------------------
__global__ void __launch_bounds__(128) out_kernel(
    const float* __restrict__ qbuf, const float* __restrict__ X,
    const float* __restrict__ kvbuf, const float* __restrict__ kmean,
    const float* __restrict__ lepe_w, const float* __restrict__ lepe_b,
    float* __restrict__ out)
{
    __shared__ float q_s[C_], qr_s[C_], km_s[C_], kv_s[NH * HD * HD];
    const int n = blockIdx.x;
    const int b = blockIdx.y;
    const int t = threadIdx.x;
    const size_t base = ((size_t)b * N_ + n) * C_;

    q_s[t]  = qbuf[base + t];
    km_s[t] = kmean[b * C_ + t];
    for (int i = t; i < NH * HD * HD; i += C_)
        kv_s[i] = kvbuf[(size_t)b * NH * HD * HD + i];
    __syncthreads();

    // RoPE on q (angle depends on w = n % 160 due to reference broadcast quirk)
    const int pair = t >> 1;
    const float theta = __powf(10000.0f, -(float)pair * (1.0f / 64.0f));
    const int ww = n % W_;
    float s, c;
    __sincosf((float)ww * theta, &s, &c);
    const float re = q_s[t & ~1], im = q_s[t | 1];
    qr_s[t] = (t & 1) ? (re * s + im * c) : (re * c - im * s);
    __syncthreads();

    const int h = t >> 4, e = t & 15;
    float zden = 1e-6f, acc = 0.f;
#pragma unroll
    for (int d = 0; d < HD; ++d) {
        zden += q_s[h * HD + d] * km_s[h * HD + d];
        acc  += qr_s[h * HD + d] * kv_s[(h * HD + d) * HD + e];
    }
    float res = acc / zden;

    // LePE: depthwise 3x3 conv on v (= raw-reshaped x), SAME padding
    const int hh = n / W_;
    float lep = lepe_b[t];
#pragma unroll
    for (int ky = 0; ky < 3; ++ky) {
        const int ih = hh + ky - 1;
        if (ih < 0 || ih >= H_) continue;
#pragma unroll
        for (int kx = 0; kx < 3; ++kx) {
            const int iw = ww + kx - 1;
            if (iw < 0 || iw >= W_) continue;
            lep += lepe_w[t * 9 + ky * 3 + kx] *
                   X[((size_t)b * N_ + ih * W_ + iw) * C_ + t];
        }
    }

    out[((size_t)b * C_ + t) * N_ + n] = res + lep;
}

// ---------------------------------------------------------------------------
__global__ void zero_kernel(float* __restrict__ p, int n)
{
    int i = blockIdx.x * blockDim.x + threadIdx.x;
    if (i < n) p[i] = 0.f;
}

extern "C" void kernel_launch(void* const* d_in, const int* in_sizes, int n_in,
                              void* d_out, int out_size, void* d_ws, size_t ws_size,
                              hipStream_t stream)
{
    const float* x      = (const float*)d_in[0];
    const float* qk_w   = (const float*)d_in[1];
    const float* qk_b   = (const float*)d_in[2];
    const float* lepe_w = (const float*)d_in[3];
    const float* lepe_b = (const float*)d_in[4];
    // d_in[5] = num_heads (fixed 8, hardcoded)

    float* qbuf  = (float*)d_ws;
    float* kbuf  = qbuf + (size_t)B_ * N_ * C_;
    float* kvbuf = kbuf + (size_t)B_ * N_ * C_;
    float* kmean = kvbuf + B_ * NH * HD * HD;  // contiguous after kv
    float* out   = (float*)d_out;

    const int nz = B_ * NH * HD * HD + B_ * C_;  // zero kv + kmean together
    zero_kernel<<<(nz + 255) / 256, 256, 0, stream>>>(kvbuf, nz);

    qk_gemm_kernel<<<B_ * (N_ / 16) * (TWO_C / 16) / 8, 256, 0, stream>>>(
        x, qk_w, qk_b, qbuf, kbuf);

    kmean_kernel<<<B_ * C_, 256, 0, stream>>>(kbuf, kmean);

    kv_kernel<<<B_ * NH * SEG, 256, 0, stream>>>(kbuf, x, kvbuf);

    out_kernel<<<dim3(N_, B_), 128, 0, stream>>>(
        qbuf, x, kvbuf, kmean, lepe_w, lepe_b, out);
}